// Light_user_layer_23493471109151
// MI455X (gfx1250) — compile-verified
//
#include <hip/hip_runtime.h>

typedef __attribute__((ext_vector_type(2))) float v2f;
typedef __attribute__((ext_vector_type(8))) float v8f;

#define DIM 64

// One wave32 processes a tile of 16 edges.
//  - lanes 0..15 own edge (tileBase + lane) metadata (row/col/val); lanes 16..31 mirror.
//  - slot[e] = first lane in tile with the same destination row (tile-local segment id).
//  - A (16 slots x 16 edges, fp32) carries vals at (slot[e], e); split into 4 K=4 chunks.
//  - B (16 edges x 64 dims) = gathered embedding rows; 4 dim-chunks of 16.
//  - D = A x B accumulated with V_WMMA_F32_16X16X4_F32 (4 K-chunks x 4 dim-chunks).
//  - Only slot owners scatter their 64-wide row with global_atomic_add_f32.
__global__ __launch_bounds__(256) void spmm_coo_wmma_kernel(
    const float* __restrict__ emb,
    const float* __restrict__ vals,
    const int*   __restrict__ rows,
    const int*   __restrict__ cols,
    float*       __restrict__ out,
    int E)
{
    const int lane  = threadIdx.x & 31;
    const int wave  = (int)((blockIdx.x * blockDim.x + threadIdx.x) >> 5);
    const int ebase = wave * 16;
    if (ebase >= E) return;                 // wave-uniform exit: EXEC stays all-1s for WMMA

    const int l15  = lane & 15;
    const int half = lane >> 4;             // 0 or 1

    // Edge metadata (lanes 16..31 duplicate lanes 0..15; same addresses -> coalesced)
    const int  eIdx = ebase + l15;
    const bool vld  = (eIdx < E);
    const int  row  = vld ? rows[eIdx] : 0;
    const int  col  = vld ? cols[eIdx] : 0;
    const float val = vld ? vals[eIdx] : 0.0f;

    // Tile-local segment id: smallest lane j (0..15) with the same row.
    int slot = l15;
    #pragma unroll
    for (int j = 14; j >= 0; --j) {
        int rj = __shfl(row, j, 32);
        if (j < l15 && rj == row) slot = j;
    }

    v8f acc0 = {}, acc1 = {}, acc2 = {}, acc3 = {};

    #pragma unroll
    for (int kc = 0; kc < 4; ++kc) {
        // K mapping for 32-bit 16x16x4 operands: K = 2*half + vgpr (same for A and B,
        // so any consistent K relabeling cancels in the product).
        const int le0 = kc * 4 + half * 2;       // local edge for vector element .x
        const int le1 = le0 + 1;                 // local edge for vector element .y

        const int   s0 = __shfl(slot, le0, 32);
        const int   s1 = __shfl(slot, le1, 32);
        const float w0 = __shfl(val,  le0, 32);
        const float w1 = __shfl(val,  le1, 32);
        v2f a;
        a.x = (s0 == l15) ? w0 : 0.0f;           // A[M=l15][K=2*half+0]
        a.y = (s1 == l15) ? w1 : 0.0f;           // A[M=l15][K=2*half+1]

        const int c0 = __shfl(col, le0, 32) * DIM;
        const int c1 = __shfl(col, le1, 32) * DIM;

        v2f b;
        b.x = emb[c0 +  0 + l15];  b.y = emb[c1 +  0 + l15];   // B[K][N=l15], dims 0..15
        acc0 = __builtin_amdgcn_wmma_f32_16x16x4_f32(false, a, false, b, (short)0, acc0, false, false);
        b.x = emb[c0 + 16 + l15];  b.y = emb[c1 + 16 + l15];   // dims 16..31
        acc1 = __builtin_amdgcn_wmma_f32_16x16x4_f32(false, a, false, b, (short)0, acc1, false, false);
        b.x = emb[c0 + 32 + l15];  b.y = emb[c1 + 32 + l15];   // dims 32..47
        acc2 = __builtin_amdgcn_wmma_f32_16x16x4_f32(false, a, false, b, (short)0, acc2, false, false);
        b.x = emb[c0 + 48 + l15];  b.y = emb[c1 + 48 + l15];   // dims 48..63
        acc3 = __builtin_amdgcn_wmma_f32_16x16x4_f32(false, a, false, b, (short)0, acc3, false, false);
    }

    // Scatter: D row M lives in acc*[g] with M = g + 8*half, N = l15 (documented C/D layout).
    // Only slot owners (first occurrence of a row in the tile) carry nonzero rows.
    #pragma unroll
    for (int g = 0; g < 8; ++g) {
        const int M  = g + half * 8;
        const int rM = __shfl(row,  M, 32);
        const int sM = __shfl(slot, M, 32);
        float* dst = out + rM * DIM + l15;
        if (sM == M) {
            atomicAdd(dst +  0, acc0[g]);
            atomicAdd(dst + 16, acc1[g]);
            atomicAdd(dst + 32, acc2[g]);
            atomicAdd(dst + 48, acc3[g]);
        }
    }
}

static inline void launch_spmm(const float* emb, const float* vals,
                               const int* rows, const int* cols,
                               float* out, int E, hipStream_t stream)
{
    const int tiles   = (E + 15) / 16;           // one wave32 per tile
    const long threads = (long)tiles * 32;
    const int blocks  = (int)((threads + 255) / 256);
    spmm_coo_wmma_kernel<<<blocks, 256, 0, stream>>>(emb, vals, rows, cols, out, E);
}

extern "C" void kernel_launch(void* const* d_in, const int* in_sizes, int n_in,
                              void* d_out, int out_size, void* d_ws, size_t ws_size,
                              hipStream_t stream)
{
    // setup_inputs() order:
    const float* users_emb = (const float*)d_in[0];   // [NUM_USER, 64]
    const float* items_emb = (const float*)d_in[1];   // [NUM_ITEM, 64]
    const float* user_vals = (const float*)d_in[2];   // [NUM_UE]
    const float* item_vals = (const float*)d_in[3];   // [NUM_IE]
    const int*   user_rows = (const int*)d_in[4];
    const int*   user_cols = (const int*)d_in[5];
    const int*   item_rows = (const int*)d_in[6];
    const int*   item_cols = (const int*)d_in[7];

    const int NUM_USER = in_sizes[0] / DIM;
    const int NUM_UE   = in_sizes[2];
    const int NUM_IE   = in_sizes[3];

    float* out_u = (float*)d_out;
    float* out_i = out_u + (size_t)NUM_USER * DIM;

    // Output is atomically accumulated; harness poisons d_out, so zero it first
    // (async memset is graph-capture safe and stream-ordered before the kernels).
    hipMemsetAsync(d_out, 0, (size_t)out_size * sizeof(float), stream);

    launch_spmm(users_emb, user_vals, user_rows, user_cols, out_u, NUM_UE, stream);
    launch_spmm(items_emb, item_vals, item_rows, item_cols, out_i, NUM_IE, stream);
}